// DetectionLoss_54666343743865
// MI455X (gfx1250) — compile-verified
//
#include <hip/hip_runtime.h>
#include <hip/hip_bf16.h>
#include <math.h>

#define IMG   640.0f
#define NCLS  80
#define NB    32
#define NG    32
#define NANCH 8400
#define BIGD  1e9f
#define PARTS_X 33            // ceil(8400/256)
#define NPART (NB * PARTS_X)  // 1056

typedef float v2f __attribute__((ext_vector_type(2)));
typedef float v8f __attribute__((ext_vector_type(8)));

// Exact f32 wave32 sum using V_WMMA_F32_16X16X4_F32 with B = ones.
// A layout (ISA 7.12.2): lane m (0..15) holds A[m, K=0..1] in its 2 VGPRs,
// lane m+16 holds A[m, K=2..3]. With A = (q, 0) per lane and B = ones:
//   D[m, n] = q_m + q_{m+16}
// C/D layout: VGPR r, lanes 0-15 -> M=r, lanes 16-31 -> M=r+8. So
// sum_r d[r] on a lane = partial row-sum over half the M's; adding the
// lane^16 partner gives the full 32-lane sum in every lane.
__device__ __forceinline__ float wave_sum32(float q) {
  v2f a;  a[0] = q;    a[1] = 0.0f;
  v2f bm; bm[0] = 1.0f; bm[1] = 1.0f;
  v8f c = {};
  v8f d = __builtin_amdgcn_wmma_f32_16x16x4_f32(
      /*neg_a=*/false, a, /*neg_b=*/false, bm,
      /*c_mod=*/(short)0, c, /*reuse_a=*/false, /*reuse_b=*/false);
  float s = d[0] + d[1] + d[2] + d[3] + d[4] + d[5] + d[6] + d[7];
  s += __shfl_xor(s, 16);
  return s;
}

__global__ void k_init(int* __restrict__ matched, int n) {
  int i = blockIdx.x * blockDim.x + threadIdx.x;
  if (i < n) matched[i] = -1;
}

// One wave per (b, g, level). Stable top-10 of masked L1 center distance,
// scatter-max gt index into matched. Ordering key = (dist, anchor_idx) lex,
// matching jax.lax.top_k stability exactly.
__global__ void k_assign(const float* __restrict__ gt_boxes,
                         int* __restrict__ matched) {
  const int   Wl[3]   = {80, 40, 20};
  const int   HWl[3]  = {6400, 1600, 400};
  const int   OFFl[3] = {0, 6400, 8000};
  const float Sl[3]   = {8.f, 16.f, 32.f};

  int t    = blockIdx.x * blockDim.x + threadIdx.x;
  int w    = t >> 5;
  int lane = t & 31;
  int b    = w / (NG * 3);
  int rem  = w - b * (NG * 3);
  int g    = rem / 3;
  int lvl  = rem - g * 3;

  const float* gb = gt_boxes + (size_t)(b * NG + g) * 4;
  float g0 = gb[0], g1 = gb[1], g2 = gb[2], g3 = gb[3];
  float s  = Sl[lvl];
  float r  = 2.5f * s;
  float gx1 = fminf(fmaxf(g0 - r, 0.f), IMG);
  float gy1 = fminf(fmaxf(g1 - r, 0.f), IMG);
  float gx2 = fminf(fmaxf(g2 + r, 0.f), IMG);
  float gy2 = fminf(fmaxf(g3 + r, 0.f), IMG);
  float gcx = (g0 + g2) * 0.5f;
  float gcy = (g1 + g3) * 0.5f;

  int W = Wl[lvl], HW = HWl[lvl];

  // Per-lane sorted top-10 by (dist, idx). Strict '<' insertion + increasing
  // idx scan order => stable tie-break (smaller idx first).
  float bd[10]; int bi[10];
#pragma unroll
  for (int j = 0; j < 10; ++j) { bd[j] = 3.4e38f; bi[j] = 0x7fffffff; }

  for (int idx = lane; idx < HW; idx += 32) {
    int y = idx / W, x = idx - y * W;
    float cx = ((float)x + 0.5f) * s;
    float cy = ((float)y + 0.5f) * s;
    bool inside = (cx >= gx1) && (cx <= gx2) && (cy >= gy1) && (cy <= gy2);
    float d  = fabsf(cx - gcx) + fabsf(cy - gcy);
    float dm = inside ? d : BIGD;
    if (dm < bd[9]) {
      bd[9] = dm; bi[9] = idx;
#pragma unroll
      for (int j = 9; j > 0; --j) {
        if (bd[j] < bd[j - 1]) {
          float td = bd[j]; bd[j] = bd[j - 1]; bd[j - 1] = td;
          int   ti = bi[j]; bi[j] = bi[j - 1]; bi[j - 1] = ti;
        }
      }
    }
  }

  // 10-round wave merge: pop the global (dist, idx)-lex minimum each round.
  int* mrow = matched + (size_t)b * NANCH + OFFl[lvl];
#pragma unroll 1
  for (int rnd = 0; rnd < 10; ++rnd) {
    float d0 = bd[0]; int i0 = bi[0];
    float dmin = d0;
#pragma unroll
    for (int o = 16; o > 0; o >>= 1) dmin = fminf(dmin, __shfl_xor(dmin, o));
    int cand = (d0 == dmin) ? i0 : 0x7fffffff;
#pragma unroll
    for (int o = 16; o > 0; o >>= 1) cand = min(cand, __shfl_xor(cand, o));
    if (d0 == dmin && i0 == cand) {  // unique winner (disjoint idx partition)
      atomicMax(&mrow[i0], g);
#pragma unroll
      for (int j = 0; j < 9; ++j) { bd[j] = bd[j + 1]; bi[j] = bi[j + 1]; }
      bd[9] = 3.4e38f; bi[9] = 0x7fffffff;
    }
  }
}

// Streaming loss kernel: grid (33, 32), block 256. One thread per anchor.
// Coalesced over hw for each class plane. WMMA wave reductions at the end
// (EXEC all-ones: no early returns, tail threads contribute zeros).
__global__ void k_loss(const float* __restrict__ cls0, const float* __restrict__ cls1,
                       const float* __restrict__ cls2, const float* __restrict__ box0,
                       const float* __restrict__ box1, const float* __restrict__ box2,
                       const float* __restrict__ gt_boxes, const int* __restrict__ gt_labels,
                       const int* __restrict__ matched,
                       float* __restrict__ cls_part, float* __restrict__ box_part,
                       float* __restrict__ pos_part) {
  int tid = threadIdx.x;
  int b   = blockIdx.y;
  int n   = blockIdx.x * blockDim.x + tid;
  bool valid = (n < NANCH);

  float cls_acc = 0.f, box_acc = 0.f, pos_acc = 0.f;

  if (valid) {
    const float* clsB; const float* boxB; int HW, W, hw; float s;
    if (n < 6400)      { clsB = cls0; boxB = box0; HW = 6400; W = 80; s = 8.f;  hw = n; }
    else if (n < 8000) { clsB = cls1; boxB = box1; HW = 1600; W = 40; s = 16.f; hw = n - 6400; }
    else               { clsB = cls2; boxB = box2; HW = 400;  W = 20; s = 32.f; hw = n - 8000; }
    int yy = hw / W, xx = hw - yy * W;
    float cx = ((float)xx + 0.5f) * s;
    float cy = ((float)yy + 0.5f) * s;

    int  m   = matched[(size_t)b * NANCH + n];
    bool pos = (m >= 0);
    int  mg  = pos ? m : 0;
    const float* gb = gt_boxes + (size_t)(b * NG + mg) * 4;
    float g0 = gb[0], g1 = gb[1], g2 = gb[2], g3 = gb[3];
    int gc = gt_labels[b * NG + mg];

    // BCE over 80 classes: max(x,0) - x*t + log1p(exp(-|x|))
    const float* cp = clsB + (size_t)b * NCLS * HW + hw;
#pragma unroll 4
    for (int c = 0; c < NCLS; ++c) {
      float x = cp[(size_t)c * HW];
      float bce = fmaxf(x, 0.f) + log1pf(expf(-fabsf(x)));
      if (pos && c == gc) bce -= x;
      cls_acc += bce;
    }

    if (pos) {  // (1-iou)*posf: identical to masked unconditional compute
      const float* bp = boxB + (size_t)b * 4 * HW + hw;
      float lp = bp[0], tp = bp[(size_t)HW], rp = bp[(size_t)2 * HW], dp = bp[(size_t)3 * HW];
      float px1 = fminf(fmaxf(cx - lp, 0.f), IMG);
      float py1 = fminf(fmaxf(cy - tp, 0.f), IMG);
      float px2 = fminf(fmaxf(cx + rp, 0.f), IMG);
      float py2 = fminf(fmaxf(cy + dp, 0.f), IMG);
      float tl = fmaxf(cx - g0, 0.f);
      float tt = fmaxf(cy - g1, 0.f);
      float tr = fmaxf(g2 - cx, 0.f);
      float tb = fmaxf(g3 - cy, 0.f);
      float tx1 = cx - tl, ty1 = cy - tt, tx2 = cx + tr, ty2 = cy + tb;
      float ix1 = fmaxf(px1, tx1), iy1 = fmaxf(py1, ty1);
      float ix2 = fminf(px2, tx2), iy2 = fminf(py2, ty2);
      float inter = fmaxf(ix2 - ix1, 0.f) * fmaxf(iy2 - iy1, 0.f);
      float ap = fmaxf(px2 - px1, 0.f) * fmaxf(py2 - py1, 0.f);
      float at = fmaxf(tx2 - tx1, 0.f) * fmaxf(ty2 - ty1, 0.f);
      float iou = inter / (ap + at - inter + 1e-6f);
      box_acc = 1.f - iou;
      pos_acc = 1.f;
    }
  }

  // WMMA wave32 reductions (all 256 threads reach here -> EXEC all ones)
  float cw = wave_sum32(cls_acc);
  float bw = wave_sum32(box_acc);
  float pw = wave_sum32(pos_acc);

  __shared__ float red[24];
  int wv = tid >> 5, lane = tid & 31;
  if (lane == 0) { red[wv] = cw; red[8 + wv] = bw; red[16 + wv] = pw; }
  __syncthreads();
  if (tid == 0) {
    float c = 0.f, bx = 0.f, p = 0.f;
#pragma unroll
    for (int i = 0; i < 8; ++i) { c += red[i]; bx += red[8 + i]; p += red[16 + i]; }
    int part = blockIdx.y * gridDim.x + blockIdx.x;
    cls_part[part] = c; box_part[part] = bx; pos_part[part] = p;
  }
}

// Single-block deterministic finalize.
__global__ void k_finalize(const float* __restrict__ cls_part,
                           const float* __restrict__ box_part,
                           const float* __restrict__ pos_part,
                           float* __restrict__ out) {
  __shared__ float sh[256];
  __shared__ float img[32];
  int tid = threadIdx.x;
  float a = 0.f;
  for (int i = tid; i < NPART; i += 256) a += cls_part[i];
  sh[tid] = a;
  __syncthreads();
  for (int st = 128; st > 0; st >>= 1) {
    if (tid < st) sh[tid] += sh[tid + st];
    __syncthreads();
  }
  if (tid < 32) {
    float bs = 0.f, ps = 0.f;
    for (int j = 0; j < PARTS_X; ++j) {
      bs += box_part[tid * PARTS_X + j];
      ps += pos_part[tid * PARTS_X + j];
    }
    img[tid] = (ps > 0.f) ? (bs / fmaxf(ps, 1.f)) : 0.f;
  }
  __syncthreads();
  if (tid == 0) {
    float tb = 0.f;
    for (int i = 0; i < 32; ++i) tb += img[i];
    out[0] = 1.0f * (sh[0] / (float)NANCH) + 2.5f * tb;
  }
}

extern "C" void kernel_launch(void* const* d_in, const int* in_sizes, int n_in,
                              void* d_out, int out_size, void* d_ws, size_t ws_size,
                              hipStream_t stream) {
  (void)in_sizes; (void)n_in; (void)out_size; (void)ws_size;
  const float* cls0 = (const float*)d_in[0];
  const float* cls1 = (const float*)d_in[1];
  const float* cls2 = (const float*)d_in[2];
  const float* box0 = (const float*)d_in[3];
  const float* box1 = (const float*)d_in[4];
  const float* box2 = (const float*)d_in[5];
  const float* gtb  = (const float*)d_in[6];
  const int*   gtl  = (const int*)d_in[7];

  int*   matched  = (int*)d_ws;
  float* cls_part = (float*)((char*)d_ws + (size_t)NB * NANCH * sizeof(int));
  float* box_part = cls_part + NPART;
  float* pos_part = box_part + NPART;
  float* out      = (float*)d_out;

  int mtot = NB * NANCH;
  k_init<<<(mtot + 255) / 256, 256, 0, stream>>>(matched, mtot);

  // 3072 waves = 98304 threads, 128-thread blocks
  k_assign<<<(NB * NG * 3 * 32) / 128, 128, 0, stream>>>(gtb, matched);

  dim3 grid(PARTS_X, NB, 1);
  k_loss<<<grid, 256, 0, stream>>>(cls0, cls1, cls2, box0, box1, box2,
                                   gtb, gtl, matched,
                                   cls_part, box_part, pos_part);

  k_finalize<<<1, 256, 0, stream>>>(cls_part, box_part, pos_part, out);
}